// AvgSelfAtt_62311385530569
// MI455X (gfx1250) — compile-verified
//
#include <hip/hip_runtime.h>

// Problem constants (from reference): B=8, S=4096, D=1024
constexpr int B_ = 8;
constexpr int S_ = 4096;
constexpr int D_ = 1024;
constexpr int CS = 256;          // sequence chunk per scan workgroup
constexpr int NC = S_ / CS;      // 16 chunks
constexpr int DT = D_ / 16;      // 64 d-tiles of width 16

typedef float v2f __attribute__((ext_vector_type(2)));
typedef float v8f __attribute__((ext_vector_type(8)));

// ---------------------------------------------------------------------------
// Phase 1: per-chunk column sums.  partial[b][c][d] = sum_{s in chunk c} x[b,s,d]
// grid = B*NC blocks, 1024 threads (one per d). Fully coalesced along d.
// ---------------------------------------------------------------------------
__global__ void __launch_bounds__(1024)
avgatt_partial_sums(const float* __restrict__ x, float* __restrict__ partial) {
    const int blk = blockIdx.x;
    const int b = blk / NC;
    const int c = blk % NC;
    const int d = threadIdx.x;            // 0..1023

    const float* xp = x + ((size_t)b * S_ + (size_t)c * CS) * D_ + d;
    float acc = 0.0f;
    #pragma unroll 8
    for (int s = 0; s < CS; ++s) {
        acc += xp[(size_t)s * D_];
    }
    partial[((size_t)b * NC + c) * D_ + d] = acc;
}

// ---------------------------------------------------------------------------
// Phase 2: blocked prefix scan via f32 WMMA.
// One wave32 per (b, chunk, 16-wide d-tile). Within the chunk, each 16x16
// (seq x d) tile is prefix-summed as  L16 @ X  using V_WMMA_F32_16X16X4_F32
// chained over 4 K-steps (exact f32 adds since L16 is 0/1).
//
// Layouts per CDNA5 ISA (wave32):
//   A (16x4 f32):  lane L -> M = L&15;  VGPR0 holds K = (L<16 ? 0 : 2),
//                                       VGPR1 holds K = (L<16 ? 1 : 3)
//   B (4x16 f32):  mirrored: VGPR0 = rows K=0 (lanes 0-15) / K=2 (lanes 16-31),
//                            VGPR1 = rows K=1 / K=3; column N = lane&15
//   C/D (16x16):   lane L -> column N = L&15; VGPR r -> row M = r + (L<16?0:8)
// ---------------------------------------------------------------------------
__global__ void __launch_bounds__(32)
avgatt_scan_wmma(const float* __restrict__ x,
                 const float* __restrict__ partial,
                 float* __restrict__ out) {
    const int lane = threadIdx.x;                 // 0..31
    const int blk  = blockIdx.x;                  // b*NC*DT + c*DT + dt
    const int dt   = blk % DT;
    const int c    = (blk / DT) % NC;
    const int b    = blk / (DT * NC);

    const int col = lane & 15;                    // d within tile / N column
    const bool hi = (lane >= 16);
    const int d0  = dt * 16;
    const int d   = d0 + col;

    // Inter-chunk carry: sum of earlier chunks' column sums for this d.
    float carry = 0.0f;
    for (int cc = 0; cc < c; ++cc) {
        carry += partial[((size_t)b * NC + cc) * D_ + d];
    }

    const float* xbase = x   + ((size_t)b * S_ + (size_t)c * CS) * D_ + d0;
    float*       obase = out + ((size_t)b * S_ + (size_t)c * CS) * D_ + d0;

    const int M_A = lane & 15;                    // A-matrix row owned by lane

    for (int t = 0; t < CS / 16; ++t) {
        const int s0 = t * 16;                    // tile base within chunk

        v8f acc = {};                             // C = 0
        #pragma unroll
        for (int k = 0; k < 4; ++k) {
            const int k0 = 4 * k + (hi ? 2 : 0);  // K index in VGPR0
            const int k1 = 4 * k + (hi ? 3 : 1);  // K index in VGPR1

            v2f a, bm;
            // A = columns [4k .. 4k+3] of lower-triangular ones (M >= K)
            a.x = (M_A >= k0) ? 1.0f : 0.0f;
            a.y = (M_A >= k1) ? 1.0f : 0.0f;
            // B = rows [4k .. 4k+3] of the 16x16 x-tile
            bm.x = xbase[(size_t)(s0 + k0) * D_ + col];
            bm.y = xbase[(size_t)(s0 + k1) * D_ + col];

            acc = __builtin_amdgcn_wmma_f32_16x16x4_f32(
                /*neg_a=*/false, a, /*neg_b=*/false, bm,
                /*c_mod=*/(short)0, acc, /*reuse_a=*/false, /*reuse_b=*/false);
        }

        // Add running carry (per-lane scalar: lane owns a fixed d column).
        #pragma unroll
        for (int r = 0; r < 8; ++r) acc[r] += carry;

        // Next carry = tile row 15 (pre-division) = VGPR7 of the hi half-wave,
        // broadcast from lane (col+16) to both half-waves.
        const float newcarry = __shfl(acc[7], col + 16, 32);

        // Scale by 1/(s+1) via hardware v_rcp_f32 (~1 ulp; divisors are small
        // integers so this is numerically indistinguishable from IEEE divide
        // at realistic tolerances, and ~7x fewer VALU ops than v_div_* chain).
        // Store: row M = r + (hi?8:0); 16 lanes per row write contiguous 64B.
        #pragma unroll
        for (int r = 0; r < 8; ++r) {
            const int srow  = s0 + r + (hi ? 8 : 0);          // s within chunk
            const int sglob = c * CS + srow;                  // global s
            const float inv = __builtin_amdgcn_rcpf((float)(sglob + 1));
            obase[(size_t)srow * D_ + col] = acc[r] * inv;
        }

        carry = newcarry;
    }
}

// ---------------------------------------------------------------------------
extern "C" void kernel_launch(void* const* d_in, const int* in_sizes, int n_in,
                              void* d_out, int out_size, void* d_ws, size_t ws_size,
                              hipStream_t stream) {
    const float* x   = (const float*)d_in[0];   // [B, S, D] f32
    float*       out = (float*)d_out;           // [B, S, D] f32
    float*       partial = (float*)d_ws;        // [B, NC, D] f32 = 512 KB

    avgatt_partial_sums<<<B_ * NC, 1024, 0, stream>>>(x, partial);
    avgatt_scan_wmma<<<B_ * NC * DT, 32, 0, stream>>>(x, partial, out);
}